// MambaEncoder_2525440770250
// MI455X (gfx1250) — compile-verified
//
#include <hip/hip_runtime.h>
#include <math.h>

// ---------------- problem constants ----------------
#define NL 4
#define DM 512
#define DS 16
#define DCONV 4
#define DI 1024          // EXPAND * DM
#define DTR 32           // ceil(DM/16)
#define BB 2
#define LL 2048
#define ROWS (BB * LL)   // 4096
#define RMS_EPS 1.1920929e-07f

// chunked scan
#define NC 16            // chunks
#define CH (LL / NC)     // 128 steps per chunk

typedef float v2f __attribute__((ext_vector_type(2)));
typedef float v8f __attribute__((ext_vector_type(8)));
typedef unsigned int u32x4 __attribute__((ext_vector_type(4)));
typedef int i32x4 __attribute__((ext_vector_type(4)));
typedef int i32x8 __attribute__((ext_vector_type(8)));

__device__ __forceinline__ float silu_f(float x) {
    return x / (1.0f + expf(-x));
}
__device__ __forceinline__ float softplus_f(float x) {
    return (x > 20.0f) ? x : log1pf(expf(x));
}
__device__ __forceinline__ v8f wmma4(v2f a, v2f b, v8f c) {
    return __builtin_amdgcn_wmma_f32_16x16x4_f32(
        false, a, false, b, (short)0, c, false, false);
}

// =====================================================================
// Tensor Data Mover: 2D tile (rows x rowlen f32, global row stride in
// elements) from global memory into LDS.  One wave issues, EXEC ignored.
// D# layout per CDNA5 ISA 8.3/8.4 (group0: count/lds/global/type,
// group1: data_size, tensor dims, tile dims, dim0 stride).
// =====================================================================
__device__ __forceinline__ void tdm_load_2d_f32(
    void* lds_dst, const float* gsrc, int rows, int rowlen, int row_stride)
{
    unsigned long long ga = (unsigned long long)(uintptr_t)gsrc;
    unsigned lds_off = (unsigned)(uintptr_t)lds_dst;  // LDS aperture: low 32 bits

    u32x4 g0;
    g0.x = 1u;                                        // count=1 (valid user D#)
    g0.y = lds_off;                                   // lds_addr
    g0.z = (unsigned)(ga & 0xffffffffu);              // global_addr[31:0]
    g0.w = (unsigned)((ga >> 32) & 0x01ffffffu)       // global_addr[56:32]
         | (2u << 30);                                // type=2 ("image")

    i32x8 g1;
    g1[0] = (int)(2u << 16);                          // data_size=2 (4 bytes)
    g1[1] = (int)(((unsigned)rowlen & 0xffffu) << 16);          // tensor_dim0 lo
    g1[2] = (int)((((unsigned)rowlen >> 16) & 0xffffu)
         | (((unsigned)rows & 0xffffu) << 16));                 // dim0 hi | dim1 lo
    g1[3] = (int)((((unsigned)rows >> 16) & 0xffffu)
         | (((unsigned)rowlen & 0xffffu) << 16));               // dim1 hi | tile_dim0
    g1[4] = (int)((unsigned)rows & 0xffffu);                    // tile_dim1
    g1[5] = (int)row_stride;                                    // tensor_dim0_stride lo32
    g1[6] = 0;
    g1[7] = 0;

    i32x4 z4 = {0, 0, 0, 0};
#if defined(__clang_major__) && (__clang_major__ >= 23)
    i32x8 z8 = {0, 0, 0, 0, 0, 0, 0, 0};
    __builtin_amdgcn_tensor_load_to_lds(g0, g1, z4, z4, z8, 0);
#else
    __builtin_amdgcn_tensor_load_to_lds(g0, g1, z4, z4, 0);
#endif
}

// =====================================================================
// WMMA fp32 GEMM:  C[M,N] = A[M,K] * W[N,K]^T   (A row-major, W row-major)
// Block = 128 threads = 4 waves in 2x2; each wave owns a 32x32 tile
// (2x2 WMMA accumulators).  Register double-buffered over K so the four
// fragment loads for step k+4 overlap the four WMMAs of step k.
// grid.x = N/64, grid.y = M/64.  mode==1: fused bias + softplus epilogue.
// =====================================================================
__global__ __launch_bounds__(128) void gemm_wmma_f32(
    const float* __restrict__ A, int lda,
    const float* __restrict__ W, int ldw,
    float* __restrict__ C, int ldc,
    int K, const float* __restrict__ bias, int mode)
{
    const int lane  = threadIdx.x & 31;
    const int wave  = threadIdx.x >> 5;
    const int l15   = lane & 15;
    const int khalf = (lane >> 4) << 1;          // 0 for lanes 0-15, 2 for 16-31
    const int tileM = blockIdx.y * 64 + (wave >> 1) * 32;
    const int tileN = blockIdx.x * 64 + (wave & 1) * 32;

    const float* Ap0 = A + (size_t)(tileM + l15) * lda + khalf;
    const float* Ap1 = Ap0 + (size_t)16 * lda;
    const float* Bp0 = W + (size_t)(tileN + l15) * ldw + khalf;
    const float* Bp1 = Bp0 + (size_t)16 * ldw;

    __builtin_prefetch(Ap0, 0, 3);
    __builtin_prefetch(Bp0, 0, 3);
    __builtin_prefetch(Bp1, 0, 3);

    v8f acc[2][2] = {};
    v2f a0 = *(const v2f*)(Ap0);
    v2f a1 = *(const v2f*)(Ap1);
    v2f b0 = *(const v2f*)(Bp0);
    v2f b1 = *(const v2f*)(Bp1);

    for (int k = 4; k < K; k += 4) {
        v2f a0n = *(const v2f*)(Ap0 + k);
        v2f a1n = *(const v2f*)(Ap1 + k);
        v2f b0n = *(const v2f*)(Bp0 + k);
        v2f b1n = *(const v2f*)(Bp1 + k);
        acc[0][0] = wmma4(a0, b0, acc[0][0]);
        acc[0][1] = wmma4(a0, b1, acc[0][1]);
        acc[1][0] = wmma4(a1, b0, acc[1][0]);
        acc[1][1] = wmma4(a1, b1, acc[1][1]);
        a0 = a0n; a1 = a1n; b0 = b0n; b1 = b1n;
    }
    acc[0][0] = wmma4(a0, b0, acc[0][0]);
    acc[0][1] = wmma4(a0, b1, acc[0][1]);
    acc[1][0] = wmma4(a1, b0, acc[1][0]);
    acc[1][1] = wmma4(a1, b1, acc[1][1]);

    // C layout: lanes 0-15 -> M = base+v, lanes 16-31 -> M = base+8+v; N = l15
    const int vrow = (lane >> 4) << 3;
#pragma unroll
    for (int mi = 0; mi < 2; ++mi) {
#pragma unroll
        for (int nj = 0; nj < 2; ++nj) {
            const int n = tileN + nj * 16 + l15;
            const float bv = (mode == 1) ? bias[n] : 0.0f;
#pragma unroll
            for (int v = 0; v < 8; ++v) {
                float val = acc[mi][nj][v];
                if (mode == 1) val = softplus_f(val + bv);
                C[(size_t)(tileM + mi * 16 + vrow + v) * ldc + n] = val;
            }
        }
    }
}

// =====================================================================
// Depthwise causal conv (K=4) + bias + SiLU.
// =====================================================================
__global__ __launch_bounds__(256) void conv_silu_kernel(
    const float* __restrict__ xz, const float* __restrict__ cw,
    const float* __restrict__ cb, float* __restrict__ xc)
{
    int idx = blockIdx.x * 256 + threadIdx.x;          // over BB*LL*DI
    if (idx >= ROWS * DI) return;
    int d = idx & (DI - 1);
    int l = (idx >> 10) & (LL - 1);
    int b = idx >> 21;

    float acc = cb[d];
#pragma unroll
    for (int k = 0; k < DCONV; ++k) {
        int lp = l - (DCONV - 1) + k;
        if (lp >= 0)
            acc += xz[(size_t)(b * LL + lp) * (2 * DI) + d] * cw[d * DCONV + k];
    }
    xc[idx] = silu_f(acc);
}

// =====================================================================
// Scan phase 1: per (b, chunk, d) compute chunk-local final state and
// per-state product of dA.  B staged in LDS via the Tensor Data Mover.
// grid.x = BB*NC*(DI/256), block = 256.
// =====================================================================
__global__ __launch_bounds__(256) void scan_phase1(
    const float* __restrict__ dt, const float* __restrict__ xc,
    const float* __restrict__ dbc, const float* __restrict__ A_log,
    float* __restrict__ hend, float* __restrict__ Pbuf)
{
    const int bid  = blockIdx.x;
    const int dgrp = bid & 3;
    const int c    = (bid >> 2) & (NC - 1);
    const int b    = bid >> 6;
    const int d    = dgrp * 256 + threadIdx.x;

    __shared__ float sB[CH * DS];
    if (threadIdx.x < 32) {   // wave 0 drives the TDM (EXEC ignored by TDM)
        tdm_load_2d_f32(sB, dbc + (size_t)(b * LL + c * CH) * 64 + DTR,
                        CH, DS, 64);
        __builtin_amdgcn_s_wait_tensorcnt(0);
    }
    __syncthreads();

    float Ar[DS], h[DS], P[DS];
#pragma unroll
    for (int s = 0; s < DS; ++s) {
        Ar[s] = -expf(A_log[d * DS + s]);
        h[s] = 0.0f; P[s] = 1.0f;
    }

    for (int t = 0; t < CH; ++t) {
        size_t ro = (size_t)(b * LL + c * CH + t) * DI + d;
        float dtv = dt[ro];
        float dx  = dtv * xc[ro];
#pragma unroll
        for (int s = 0; s < DS; ++s) {
            float dA = expf(dtv * Ar[s]);
            h[s] = h[s] * dA + dx * sB[t * DS + s];
            P[s] *= dA;
        }
    }

    size_t o = ((size_t)((b * NC + c) * DI + d)) * DS;
#pragma unroll
    for (int s = 0; s < DS; ++s) { hend[o + s] = h[s]; Pbuf[o + s] = P[s]; }
}

// =====================================================================
// Scan phase 2: serial prefix over the 16 chunks for each (b,d,s).
// =====================================================================
__global__ __launch_bounds__(256) void scan_phase2(
    const float* __restrict__ hend, const float* __restrict__ P,
    float* __restrict__ hinit)
{
    int tid = blockIdx.x * 256 + threadIdx.x;   // BB*DI*DS = 32768
    int s = tid & 15;
    int d = (tid >> 4) & (DI - 1);
    int b = tid >> 14;
    float H = 0.0f;
#pragma unroll
    for (int c = 0; c < NC; ++c) {
        size_t o = ((size_t)((b * NC + c) * DI + d)) * DS + s;
        hinit[o] = H;
        H = H * P[o] + hend[o];
    }
}

// =====================================================================
// Scan phase 3: replay each chunk with correct h0, emit y (+ D skip).
// B and C tiles staged in LDS via two TDM descriptors.
// =====================================================================
__global__ __launch_bounds__(256) void scan_phase3(
    const float* __restrict__ dt, const float* __restrict__ xc,
    const float* __restrict__ dbc, const float* __restrict__ A_log,
    const float* __restrict__ D_skip, const float* __restrict__ hinit,
    float* __restrict__ y)
{
    const int bid  = blockIdx.x;
    const int dgrp = bid & 3;
    const int c    = (bid >> 2) & (NC - 1);
    const int b    = bid >> 6;
    const int d    = dgrp * 256 + threadIdx.x;

    __shared__ float sB[CH * DS];
    __shared__ float sC[CH * DS];
    if (threadIdx.x < 32) {
        const float* base = dbc + (size_t)(b * LL + c * CH) * 64 + DTR;
        tdm_load_2d_f32(sB, base,      CH, DS, 64);
        tdm_load_2d_f32(sC, base + DS, CH, DS, 64);
        __builtin_amdgcn_s_wait_tensorcnt(0);
    }
    __syncthreads();

    float Ar[DS], h[DS];
    size_t ho = ((size_t)((b * NC + c) * DI + d)) * DS;
#pragma unroll
    for (int s = 0; s < DS; ++s) {
        Ar[s] = -expf(A_log[d * DS + s]);
        h[s] = hinit[ho + s];
    }
    const float Dk = D_skip[d];

    for (int t = 0; t < CH; ++t) {
        size_t ro = (size_t)(b * LL + c * CH + t) * DI + d;
        float dtv = dt[ro];
        float xv  = xc[ro];
        float dx  = dtv * xv;
        float yv = 0.0f;
#pragma unroll
        for (int s = 0; s < DS; ++s) {
            float dA = expf(dtv * Ar[s]);
            h[s] = h[s] * dA + dx * sB[t * DS + s];
            yv += h[s] * sC[t * DS + s];
        }
        y[ro] = yv + xv * Dk;
    }
}

// =====================================================================
// Gate: y *= silu(z); z = second half of xz row.
// =====================================================================
__global__ __launch_bounds__(256) void gate_kernel(
    float* __restrict__ y, const float* __restrict__ xz)
{
    int idx = blockIdx.x * 256 + threadIdx.x;
    if (idx >= ROWS * DI) return;
    int d = idx & (DI - 1);
    int row = idx >> 10;
    float z = xz[(size_t)row * (2 * DI) + DI + d];
    y[idx] *= silu_f(z);
}

// =====================================================================
// Residual add + RMSNorm, one block (256 thr) per row of 512.
// =====================================================================
__global__ __launch_bounds__(256) void addnorm_kernel(
    const float* __restrict__ proj, const float* __restrict__ res,
    const float* __restrict__ w, float* __restrict__ out)
{
    const int row = blockIdx.x;
    const int tid = threadIdx.x;
    size_t base = (size_t)row * DM;
    float v0 = proj[base + tid]       + res[base + tid];
    float v1 = proj[base + 256 + tid] + res[base + 256 + tid];

    __shared__ float sred[256];
    sred[tid] = v0 * v0 + v1 * v1;
    __syncthreads();
    for (int off = 128; off > 0; off >>= 1) {
        if (tid < off) sred[tid] += sred[tid + off];
        __syncthreads();
    }
    float scale = rsqrtf(sred[0] / (float)DM + RMS_EPS);
    out[base + tid]       = v0 * scale * w[tid];
    out[base + 256 + tid] = v1 * scale * w[tid + 256];
}

// =====================================================================
// Host side
// =====================================================================
extern "C" void kernel_launch(void* const* d_in, const int* in_sizes, int n_in,
                              void* d_out, int out_size, void* d_ws, size_t ws_size,
                              hipStream_t stream) {
    (void)in_sizes; (void)n_in; (void)out_size; (void)ws_size;

    const float* x_in    = (const float*)d_in[0];
    const float* in_w    = (const float*)d_in[1];
    const float* conv_w  = (const float*)d_in[2];
    const float* conv_b  = (const float*)d_in[3];
    const float* xp_w    = (const float*)d_in[4];
    const float* dtp_w   = (const float*)d_in[5];
    const float* dtp_b   = (const float*)d_in[6];
    const float* A_log   = (const float*)d_in[7];
    const float* D_skip  = (const float*)d_in[8];
    const float* out_w   = (const float*)d_in[9];
    const float* rms_w   = (const float*)d_in[10];

    float* ws = (float*)d_ws;
    float* buf_xz    = ws;                                     // 4096*2048
    float* buf_xc    = buf_xz    + (size_t)ROWS * 2 * DI;      // 4096*1024
    float* buf_dbc   = buf_xc    + (size_t)ROWS * DI;          // 4096*64
    float* buf_dt    = buf_dbc   + (size_t)ROWS * 64;          // 4096*1024
    float* buf_y     = buf_dt    + (size_t)ROWS * DI;          // 4096*1024
    float* buf_hend  = buf_y     + (size_t)ROWS * DI;          // 2*16*1024*16
    float* buf_P     = buf_hend  + (size_t)BB * NC * DI * DS;
    float* buf_hinit = buf_P     + (size_t)BB * NC * DI * DS;
    float* buf_xcur  = buf_hinit + (size_t)BB * NC * DI * DS;  // 4096*512
    float* buf_proj  = buf_xcur  + (size_t)ROWS * DM;          // 4096*512

    for (int layer = 0; layer < NL; ++layer) {
        const float* xcur = (layer == 0) ? x_in : buf_xcur;
        const float* w_in  = in_w  + (size_t)layer * 2 * DI * DM;
        const float* w_cv  = conv_w + (size_t)layer * DI * DCONV;
        const float* b_cv  = conv_b + (size_t)layer * DI;
        const float* w_xp  = xp_w  + (size_t)layer * (DTR + 2 * DS) * DI;
        const float* w_dt  = dtp_w + (size_t)layer * DI * DTR;
        const float* b_dt  = dtp_b + (size_t)layer * DI;
        const float* a_log = A_log + (size_t)layer * DI * DS;
        const float* d_sk  = D_skip + (size_t)layer * DI;
        const float* w_out = out_w + (size_t)layer * DM * DI;

        // 1) in_proj: xz[4096,2048] = x[4096,512] * in_w^T
        gemm_wmma_f32<<<dim3(2 * DI / 64, ROWS / 64), 128, 0, stream>>>(
            xcur, DM, w_in, DM, buf_xz, 2 * DI, DM, nullptr, 0);

        // 2) depthwise conv + silu -> xc
        conv_silu_kernel<<<(ROWS * DI) / 256, 256, 0, stream>>>(
            buf_xz, w_cv, b_cv, buf_xc);

        // 3) x_proj: dbc[4096,64] = xc * xp_w^T
        gemm_wmma_f32<<<dim3(1, ROWS / 64), 128, 0, stream>>>(
            buf_xc, DI, w_xp, DI, buf_dbc, 64, DI, nullptr, 0);

        // 4) dt_proj + bias + softplus: dt[4096,1024]
        gemm_wmma_f32<<<dim3(DI / 64, ROWS / 64), 128, 0, stream>>>(
            buf_dbc, 64, w_dt, DTR, buf_dt, DI, DTR, b_dt, 1);

        // 5-7) chunked selective scan
        scan_phase1<<<BB * NC * (DI / 256), 256, 0, stream>>>(
            buf_dt, buf_xc, buf_dbc, a_log, buf_hend, buf_P);
        scan_phase2<<<(BB * DI * DS) / 256, 256, 0, stream>>>(
            buf_hend, buf_P, buf_hinit);
        scan_phase3<<<BB * NC * (DI / 256), 256, 0, stream>>>(
            buf_dt, buf_xc, buf_dbc, a_log, d_sk, buf_hinit, buf_y);

        // 8) gate: y *= silu(z)
        gate_kernel<<<(ROWS * DI) / 256, 256, 0, stream>>>(buf_y, buf_xz);

        // 9) out_proj: proj[4096,512] = y * out_w^T
        gemm_wmma_f32<<<dim3(DM / 64, ROWS / 64), 128, 0, stream>>>(
            buf_y, DI, w_out, DI, buf_proj, DM, DI, nullptr, 0);

        // 10) residual + RMSNorm
        float* xnext = (layer == NL - 1) ? (float*)d_out : buf_xcur;
        addnorm_kernel<<<ROWS, 256, 0, stream>>>(buf_proj, xcur, rms_w, xnext);
    }
}